// PerformerSelfAttention_3040836845895
// MI455X (gfx1250) — compile-verified
//
#include <hip/hip_runtime.h>
#include <hip/hip_bf16.h>

// Problem constants (from reference setup_inputs)
#define BB 4
#define HH 8
#define NN 8192
#define DD 64
#define MM 256
#define BH (BB*HH)

#define NORM  0.35355339059327373f   // 64^-0.25
#define NORM2 0.125f                 // 64^-0.5
#define RATIO 0.0625f                // 256^-0.5
#define FEPS  1e-4f

typedef float v2f __attribute__((ext_vector_type(2)));
typedef float v8f __attribute__((ext_vector_type(8)));

__device__ __forceinline__ v8f wmma4(v2f a, v2f b, v8f c) {
    // D(16x16,f32) = A(16x4,f32) * B(4x16,f32) + C
    return __builtin_amdgcn_wmma_f32_16x16x4_f32(
        false, a, false, b, (short)0, c, false, false);
}

// Async DMA: 16 contiguous bytes global -> 16 contiguous bytes LDS (per lane).
// GVS mode: saddr = uniform 64-bit base (SGPR pair), vaddr = per-lane byte offset.
// LDS address = low 32 bits of the flat shared pointer (workgroup-relative).
#define ASYNC_B128(ldsp, goff, base)                                          \
    asm volatile("global_load_async_to_lds_b128 %0, %1, %2 offset:0"          \
                 :: "v"((unsigned)(uintptr_t)(ldsp)), "v"((unsigned)(goff)),  \
                    "s"(base) : "memory")
#define WAIT_ASYNC() asm volatile("s_wait_asynccnt 0x0" ::: "memory")

// A-frag (16x4 f32): lanes 0-15 rows M=0..15 K={0,1}; lanes 16-31 same rows K={2,3}
__device__ __forceinline__ v2f a_frag(const float* x, int stride, int r0, int s, int lane) {
    int r = r0 + (lane & 15);
    int d = 4 * s + ((lane >> 4) << 1);
    v2f a; a.x = x[r * stride + d]; a.y = x[r * stride + d + 1];
    return a;
}

// B-frag (4x16 f32) of (NORM*proj)^T (K=d, N=m): v0 rows K={0,2}, v1 rows K={1,3}
__device__ __forceinline__ v2f projb_g(const float* proj, int mt, int s, int lane) {
    int m = mt * 16 + (lane & 15);
    int d = 4 * s + ((lane >> 4) << 1);
    v2f b; b.x = proj[m * DD + d] * NORM; b.y = proj[m * DD + d + 1] * NORM;
    return b;
}
__device__ __forceinline__ v2f projb_l(const float* pj, int mt, int s, int lane) {
    int m = mt * 16 + (lane & 15);
    int d = 4 * s + ((lane >> 4) << 1);
    v2f b; b.x = pj[m * 68 + d]; b.y = pj[m * 68 + d + 1];
    return b;
}

__device__ __forceinline__ void atomicMaxF(float* addr, float v) {
    // ordered-int trick (valid for IEEE f32, kmax init = -3e38)
    int vi = __float_as_int(v);
    if (vi >= 0) atomicMax((int*)addr, vi);
    else         atomicMin((unsigned int*)addr, (unsigned int)vi);
}

// ---------------------------------------------------------------------------
// K0: init workspace
// ---------------------------------------------------------------------------
__global__ void k0_init(float* kmax, float* ksum, float* ctx) {
    int i = blockIdx.x * 256 + threadIdx.x;           // 2048*256 == 524288 == BH*M*D
    ctx[i] = 0.0f;
    if (i < BH * MM) ksum[i] = 0.0f;
    if (i < BH) kmax[i] = -3.0e38f;
}

// ---------------------------------------------------------------------------
// K1: kmax[bh] = max over (N,M) of (norm * K * mask) @ proj^T
// grid (N/128, BH), block 256 (8 waves, 2 m-tiles each). K tile DMA'd to LDS.
// ---------------------------------------------------------------------------
__global__ __launch_bounds__(256) void k1_kmax(
    const float* __restrict__ Kg, const float* __restrict__ mask,
    const float* __restrict__ proj, float* __restrict__ kmax) {
    __shared__ __align__(16) float xn[128 * 68];   // raw K tile
    __shared__ float maskl[128];
    __shared__ float wmax[8];
    int bh = blockIdx.y;
    int b  = bh >> 3;
    long nbase = (long)blockIdx.x * 128;
    const float* Kb = Kg + ((long)bh * NN + nbase) * DD;

    for (int i = threadIdx.x; i < 128 * 16; i += 256) {   // 16-B chunks
        int r = i >> 4, wi = i & 15;
        ASYNC_B128(&xn[r * 68 + wi * 4], (r * 64 + wi * 4) * 4, Kb);
    }
    if (threadIdx.x < 128) maskl[threadIdx.x] = mask[b * NN + nbase + threadIdx.x];
    WAIT_ASYNC();
    __syncthreads();

    int lane = threadIdx.x & 31, w = threadIdx.x >> 5;
    int hf = lane >> 4;
    float mx = -3.0e38f;
    for (int mi = 0; mi < 2; ++mi) {
        int mt = w * 2 + mi;
        v2f bf[16];
        for (int s = 0; s < 16; ++s) bf[s] = projb_g(proj, mt, s, lane);
        for (int ns = 0; ns < 8; ++ns) {
            v8f c = {};
            for (int s = 0; s < 16; ++s)
                c = wmma4(a_frag(xn, 68, ns * 16, s, lane), bf[s], c);
            for (int v = 0; v < 8; ++v) {
                int r = ns * 16 + v + 8 * hf;
                mx = fmaxf(mx, c[v] * maskl[r]);   // mask is linear in K
            }
        }
    }
    for (int off = 16; off; off >>= 1) mx = fmaxf(mx, __shfl_xor(mx, off, 32));
    if (lane == 0) wmax[w] = mx;
    __syncthreads();
    if (threadIdx.x == 0) {
        float m2 = wmax[0];
        for (int i = 1; i < 8; ++i) m2 = fmaxf(m2, wmax[i]);
        atomicMaxF(&kmax[bh], m2);
    }
}

// ---------------------------------------------------------------------------
// K2: k-features -> ksum[M], ctx[M,D] += k^T @ (V*mask)
// grid (8, BH), block 512 (16 waves, wave w owns m-tile w)
// K/V chunks double-buffered through LDS with async DMA one chunk ahead.
// ---------------------------------------------------------------------------
__global__ __launch_bounds__(512) void k2_ctx(
    const float* __restrict__ Kg, const float* __restrict__ Vg,
    const float* __restrict__ mask, const float* __restrict__ proj,
    const float* __restrict__ kmax, float* __restrict__ ksum,
    float* __restrict__ ctx) {
    __shared__ __align__(16) float xnb[2][64 * 68];   // raw K
    __shared__ __align__(16) float vvb[2][64 * 68];   // raw V
    __shared__ float diag[64];                        // 0.5*NORM^2*sum(K^2) (unmasked)
    __shared__ float maskl[64];
    __shared__ float ksl[256];
    __shared__ float stg[16][16 * 17];

    int bh = blockIdx.y, b = bh >> 3;
    int lane = threadIdx.x & 31, w = threadIdx.x >> 5;
    int hf = lane >> 4;
    float km = kmax[bh];
    long rowbase = (long)blockIdx.x * 1024;

    if (threadIdx.x < 256) ksl[threadIdx.x] = 0.0f;

    auto issue = [&](int ch) {
        long nb = rowbase + (long)ch * 64;
        const float* Kb = Kg + ((long)bh * NN + nb) * DD;
        const float* Vb = Vg + ((long)bh * NN + nb) * DD;
        float* xb = xnb[ch & 1];
        float* vb = vvb[ch & 1];
        for (int i = threadIdx.x; i < 64 * 16; i += 512) {
            int r = i >> 4, wi = i & 15;
            unsigned go = (unsigned)((r * 64 + wi * 4) * 4);
            ASYNC_B128(&xb[r * 68 + wi * 4], go, Kb);
            ASYNC_B128(&vb[r * 68 + wi * 4], go, Vb);
        }
    };

    v2f bf[16];
    for (int s = 0; s < 16; ++s) bf[s] = projb_g(proj, w, s, lane);
    v8f acc[4] = {};

    issue(0);
    for (int ch = 0; ch < 16; ++ch) {
        const float* xn = xnb[ch & 1];
        const float* vv = vvb[ch & 1];
        long nb = rowbase + (long)ch * 64;
        WAIT_ASYNC();          // my chunk-ch DMA done (in-order)
        __syncthreads();       // everyone's DMA done; prev readers of other buf done
        if (ch < 15) issue(ch + 1);   // overlap next chunk's DMA with compute
        if (threadIdx.x < 64) {
            maskl[threadIdx.x] = mask[b * NN + nb + threadIdx.x];
            float sacc = 0.0f;
            for (int d = 0; d < 64; ++d) {
                float t = xn[threadIdx.x * 68 + d]; sacc += t * t;
            }
            diag[threadIdx.x] = 0.5f * NORM2 * sacc;
        }
        __syncthreads();

        for (int ns = 0; ns < 4; ++ns) {
            v8f c = {};
            for (int s = 0; s < 16; ++s)
                c = wmma4(a_frag(xn, 68, ns * 16, s, lane), bf[s], c);
            // feature: ratio*(exp(mask*xd - mask^2*diag - kmax) + eps); stage transposed
            float fsum = 0.0f;
            for (int v = 0; v < 8; ++v) {
                int r = ns * 16 + v + 8 * hf;
                float mv = maskl[r];
                float f = RATIO * (__expf(c[v] * mv - mv * mv * diag[r] - km) + FEPS);
                fsum += f;
                stg[w][(v + 8 * hf) * 17 + (lane & 15)] = f;
            }
            fsum += __shfl_xor(fsum, 16, 32);
            if (hf == 0) ksl[w * 16 + lane] += fsum;   // wave-owned, no race
            // ctx += f^T (m x n) @ (V*mask) (n x d)
            for (int kk = 0; kk < 4; ++kk) {
                int nl = kk * 4 + hf * 2;
                v2f a;
                a.x = stg[w][nl * 17 + (lane & 15)];
                a.y = stg[w][(nl + 1) * 17 + (lane & 15)];
                int nr = ns * 16 + nl;
                float m0 = maskl[nr], m1 = maskl[nr + 1];
                for (int dt = 0; dt < 4; ++dt) {
                    v2f bv;
                    bv.x = vv[nr * 68 + dt * 16 + (lane & 15)] * m0;
                    bv.y = vv[(nr + 1) * 68 + dt * 16 + (lane & 15)] * m1;
                    acc[dt] = wmma4(a, bv, acc[dt]);
                }
            }
        }
    }
    __syncthreads();
    // flush partials
    for (int dt = 0; dt < 4; ++dt)
        for (int v = 0; v < 8; ++v) {
            int m = w * 16 + v + 8 * hf;
            int d = dt * 16 + (lane & 15);
            atomicAdd(&ctx[(long)bh * MM * DD + m * DD + d], acc[dt][v]);
        }
    if (hf == 0)
        atomicAdd(&ksum[bh * MM + w * 16 + lane], ksl[w * 16 + lane]);
}

// ---------------------------------------------------------------------------
// K3: q-features (per-row max), d_inv, out = (q @ ctx) * d_inv
// grid (N/128, BH), block 256 (8 waves, wave w owns rows w*16..w*16+15)
// Q tile DMA'd raw to LDS; NORM folded into LDS-resident proj.
// ---------------------------------------------------------------------------
__global__ __launch_bounds__(256) void k3_out(
    const float* __restrict__ Qg, const float* __restrict__ proj,
    const float* __restrict__ ksum, const float* __restrict__ ctx,
    float* __restrict__ out) {
    __shared__ __align__(16) float qn[128 * 68];   // raw Q tile
    __shared__ float pj[256 * 68];                 // NORM * proj
    __shared__ float diag[128];
    __shared__ float ksl[256];
    __shared__ float stg[8][16 * 17];

    int bh = blockIdx.y;
    long nbase = (long)blockIdx.x * 128;
    int lane = threadIdx.x & 31, w = threadIdx.x >> 5;
    int hf = lane >> 4;
    const float* Qb = Qg + ((long)bh * NN + nbase) * DD;
    const float* ctxb = ctx + (long)bh * MM * DD;

    for (int i = threadIdx.x; i < 128 * 16; i += 256) {   // async Q tile
        int r = i >> 4, wi = i & 15;
        ASYNC_B128(&qn[r * 68 + wi * 4], (r * 64 + wi * 4) * 4, Qb);
    }
    for (int i = threadIdx.x; i < 256 * 64; i += 256) {
        int r = i >> 6, d = i & 63;
        pj[r * 68 + d] = proj[r * DD + d] * NORM;
    }
    if (threadIdx.x < 256) ksl[threadIdx.x] = ksum[bh * MM + threadIdx.x];
    WAIT_ASYNC();
    __syncthreads();
    if (threadIdx.x < 128) {
        float sacc = 0.0f;
        for (int d = 0; d < 64; ++d) {
            float t = qn[threadIdx.x * 68 + d]; sacc += t * t;
        }
        diag[threadIdx.x] = 0.5f * NORM2 * sacc;
    }
    __syncthreads();

    // pass 1: per-row max over all M (lane's 8 C-values are 8 rows of its half)
    float rm[8];
    for (int v = 0; v < 8; ++v) rm[v] = -3.0e38f;
    for (int mt = 0; mt < 16; ++mt) {
        v8f c = {};
        for (int s = 0; s < 16; ++s)
            c = wmma4(a_frag(qn, 68, w * 16, s, lane), projb_l(pj, mt, s, lane), c);
        for (int v = 0; v < 8; ++v) rm[v] = fmaxf(rm[v], c[v]);
    }
    for (int v = 0; v < 8; ++v)
        for (int off = 1; off < 16; off <<= 1)
            rm[v] = fmaxf(rm[v], __shfl_xor(rm[v], off, 32));

    // pass 2: features + denominator + out GEMM (ctx streamed from L2)
    v8f acc[4] = {};
    float dsum[8] = {0, 0, 0, 0, 0, 0, 0, 0};
    for (int mt = 0; mt < 16; ++mt) {
        v8f c = {};
        for (int s = 0; s < 16; ++s)
            c = wmma4(a_frag(qn, 68, w * 16, s, lane), projb_l(pj, mt, s, lane), c);
        float ks = ksl[mt * 16 + (lane & 15)];
        for (int v = 0; v < 8; ++v) {
            int r = w * 16 + v + 8 * hf;
            float f = RATIO * (__expf(c[v] - diag[r] - rm[v]) + FEPS);
            dsum[v] += f * ks;
            stg[w][(v + 8 * hf) * 17 + (lane & 15)] = f;
        }
        for (int kk = 0; kk < 4; ++kk) {
            int ml = kk * 4 + hf * 2;
            v2f a;
            a.x = stg[w][ml * 17 + (lane & 15)];
            a.y = stg[w][(ml + 1) * 17 + (lane & 15)];
            for (int dt = 0; dt < 4; ++dt) {
                int m = mt * 16 + ml;
                v2f bv;
                bv.x = ctxb[m * DD + dt * 16 + (lane & 15)];
                bv.y = ctxb[(m + 1) * DD + dt * 16 + (lane & 15)];
                acc[dt] = wmma4(a, bv, acc[dt]);
            }
        }
    }
    for (int v = 0; v < 8; ++v)
        for (int off = 1; off < 16; off <<= 1)
            dsum[v] += __shfl_xor(dsum[v], off, 32);

    for (int v = 0; v < 8; ++v) {
        float dinv = 1.0f / dsum[v];
        long row = nbase + w * 16 + v + 8 * hf;
        for (int dt = 0; dt < 4; ++dt)
            out[((long)bh * NN + row) * DD + dt * 16 + (lane & 15)] = acc[dt][v] * dinv;
    }
}

// ---------------------------------------------------------------------------
extern "C" void kernel_launch(void* const* d_in, const int* in_sizes, int n_in,
                              void* d_out, int out_size, void* d_ws, size_t ws_size,
                              hipStream_t stream) {
    (void)in_sizes; (void)n_in; (void)out_size; (void)ws_size;
    const float* Q    = (const float*)d_in[0];
    const float* K    = (const float*)d_in[1];
    const float* V    = (const float*)d_in[2];
    const float* mask = (const float*)d_in[3];
    const float* proj = (const float*)d_in[4];
    float* out  = (float*)d_out;
    float* kmax = (float*)d_ws;          // BH floats (padded to 64)
    float* ksum = kmax + 64;             // BH*M floats
    float* ctx  = ksum + BH * MM;        // BH*M*D floats

    k0_init<<<2048, 256, 0, stream>>>(kmax, ksum, ctx);
    k1_kmax<<<dim3(NN / 128, BH), 256, 0, stream>>>(K, mask, proj, kmax);
    k2_ctx <<<dim3(8, BH),        512, 0, stream>>>(K, V, mask, proj, kmax, ksum, ctx);
    k3_out <<<dim3(NN / 128, BH), 256, 0, stream>>>(Q, proj, ksum, ctx, out);
}